// AutoShot_33749853012752
// MI455X (gfx1250) — compile-verified
//
#include <hip/hip_runtime.h>

// Problem constants (fixed by setup_inputs in the reference)
#define BB 4
#define TT 100
#define TPAD 112            // 7 * 16, zero-padded rows/cols for WMMA tiles
#define HW (224 * 224)      // 50176
#define NBINS 512
#define LW 101
#define KPAD 104            // 26 * 4, zero-padded K for the head GEMM
#define PADW 50
#define OUTF 128

typedef __attribute__((ext_vector_type(2))) float v2f;
typedef __attribute__((ext_vector_type(8))) float v8f;

// ---------------------------------------------------------------------------
// Kernel 0: zero the padding rows of x ([4,112,512], rows 100..111 per batch).
// ---------------------------------------------------------------------------
__global__ __launch_bounds__(256) void xpad_kernel(float* __restrict__ x) {
  const int i = blockIdx.x * 256 + threadIdx.x;      // 4*12*512 = 24576 elems
  if (i < BB * (TPAD - TT) * NBINS) {
    const int b   = i / ((TPAD - TT) * NBINS);
    const int rem = i % ((TPAD - TT) * NBINS);
    x[((size_t)b * TPAD + TT) * NBINS + rem] = 0.f;
  }
}

// ---------------------------------------------------------------------------
// Kernel 1: per-frame 512-bin color histogram + L2 normalization.
// One workgroup (256 threads = 8 waves) per frame. Per-wave LDS sub-histograms
// reduce atomic contention; int4 loads keep the 241 MB frame read at peak BW.
// This stage is the roofline bottleneck (~10 us at 23.3 TB/s).
// ---------------------------------------------------------------------------
__global__ __launch_bounds__(256) void hist_kernel(const int* __restrict__ frames,
                                                   float* __restrict__ x) {
  __shared__ unsigned sub[8][NBINS];   // 16 KB: one sub-histogram per wave
  __shared__ float red[256];

  const int tid  = threadIdx.x;
  const int f    = blockIdx.x;         // frame id in [0, 400)
  const int b    = f / TT;
  const int t    = f % TT;
  const int wave = tid >> 5;

  for (int i = tid; i < 8 * NBINS; i += 256) ((unsigned*)sub)[i] = 0u;
  __syncthreads();

  const int* Rp = frames + ((size_t)(b * 3 + 0) * TT + t) * HW;
  const int* Gp = frames + ((size_t)(b * 3 + 1) * TT + t) * HW;
  const int* Bp = frames + ((size_t)(b * 3 + 2) * TT + t) * HW;

  // HW = 50176 = 49 * 1024; each thread does 49 int4 loads per channel plane.
  for (int it = 0; it < HW / 1024; ++it) {
    const int base = (it * 256 + tid) * 4;
    const int4 r4 = *(const int4*)(Rp + base);
    const int4 g4 = *(const int4*)(Gp + base);
    const int4 b4 = *(const int4*)(Bp + base);
    const int bin0 = ((r4.x >> 5) << 6) + ((g4.x >> 5) << 3) + (b4.x >> 5);
    const int bin1 = ((r4.y >> 5) << 6) + ((g4.y >> 5) << 3) + (b4.y >> 5);
    const int bin2 = ((r4.z >> 5) << 6) + ((g4.z >> 5) << 3) + (b4.z >> 5);
    const int bin3 = ((r4.w >> 5) << 6) + ((g4.w >> 5) << 3) + (b4.w >> 5);
    atomicAdd(&sub[wave][bin0], 1u);
    atomicAdd(&sub[wave][bin1], 1u);
    atomicAdd(&sub[wave][bin2], 1u);
    atomicAdd(&sub[wave][bin3], 1u);
  }
  __syncthreads();

  // Combine sub-histograms: thread owns bins tid and tid+256.
  unsigned c0 = 0u, c1 = 0u;
  for (int s = 0; s < 8; ++s) {
    c0 += sub[s][tid];
    c1 += sub[s][tid + 256];
  }
  const float f0 = (float)c0, f1 = (float)c1;

  // Block reduction for sum of squares (L2 norm).
  red[tid] = f0 * f0 + f1 * f1;
  __syncthreads();
  for (int s = 128; s > 0; s >>= 1) {
    if (tid < s) red[tid] += red[tid + s];
    __syncthreads();
  }
  const float inv = rsqrtf(red[0]);   // 50176 pixels -> norm > 0 always

  x[((size_t)b * TPAD + t) * NBINS + tid]       = f0 * inv;
  x[((size_t)b * TPAD + t) * NBINS + tid + 256] = f1 * inv;
}

// ---------------------------------------------------------------------------
// Kernel 2: sim[b] = X X^T per batch via V_WMMA_F32_16X16X4_F32.
// x is zero-padded to 112 rows -> branch-free inner loop (2 global_load_b64 +
// 1 wmma per K step, 128 steps, unrolled 8x / software-pipelined).
// sim is stored padded [4,112,112] -> branch-free epilogue (8 unconditional
// b32 stores with immediate offsets; the pad region is never read).
// Grid: (7 M-tiles, 4 batches); block = 7 waves, one 16x16 N-tile per wave.
// ---------------------------------------------------------------------------
__global__ __launch_bounds__(224) void sim_kernel(const float* __restrict__ x,
                                                  float* __restrict__ sim) {
  const int b     = blockIdx.y;
  const int m0    = blockIdx.x * 16;
  const int wave  = threadIdx.x >> 5;
  const int lane  = threadIdx.x & 31;
  const int n0    = wave * 16;
  const int lrow  = lane & 15;
  const int khalf = (lane >> 4) * 2;   // 0 or 2

  const int am = m0 + lrow;            // A-matrix row (M), < 112
  const int bn = n0 + lrow;            // B-matrix column (N) = row of X, < 112
  const float* xa = x + (size_t)(b * TPAD + am) * NBINS + khalf;
  const float* xb = x + (size_t)(b * TPAD + bn) * NBINS + khalf;

  v8f c = {};
#pragma unroll 8
  for (int k = 0; k < NBINS; k += 4) {
    const v2f a = *(const v2f*)(xa + k);
    const v2f w = *(const v2f*)(xb + k);
    c = __builtin_amdgcn_wmma_f32_16x16x4_f32(false, a, false, w,
                                              (short)0, c, false, false);
  }

  // C/D layout: VGPR i -> (M = m0 + i + 8*(lane>=16), N = n0 + (lane&15)).
  const int nn    = n0 + lrow;
  const int mbase = m0 + ((lane >> 4) * 8);
  float* dst = sim + ((size_t)b * TPAD + mbase) * TPAD + nn;
#pragma unroll
  for (int i = 0; i < 8; ++i) dst[i * TPAD] = c[i];
}

// ---------------------------------------------------------------------------
// Kernel 3: materialize zero-padded win [400,104] (window gather from padded
// sim) and Wpad [128,104] (zero-padded weights). One thread per element.
// ---------------------------------------------------------------------------
#define NWIN (BB * TT * KPAD)          // 41600
#define NWP  (OUTF * KPAD)             // 13312
__global__ __launch_bounds__(256) void prep_kernel(const float* __restrict__ sim,
                                                   const float* __restrict__ Wm,
                                                   float* __restrict__ win,
                                                   float* __restrict__ wpad) {
  const int i = blockIdx.x * 256 + threadIdx.x;
  if (i < NWIN) {
    const int r = i / KPAD, k = i % KPAD;
    const int b = r / TT, t = r % TT;
    const int s = t + k - PADW;
    float v = 0.f;
    if (k < LW && (unsigned)s < (unsigned)TT)
      v = sim[((size_t)b * TPAD + t) * TPAD + s];
    win[i] = v;
  } else if (i < NWIN + NWP) {
    const int j = i - NWIN;
    const int o = j / KPAD, k = j % KPAD;
    wpad[j] = (k < LW) ? Wm[(size_t)o * LW + k] : 0.f;
  }
}

// ---------------------------------------------------------------------------
// Kernel 4: out = relu(win @ Wpad^T + b). GEMM 400x104x128 via f32 WMMA;
// branch-free inner loop (operands pre-padded) and epilogue (all rows/cols
// in-bounds). Grid: 25 M-tile blocks x 8 waves (N tiles); bias + ReLU fused.
// ---------------------------------------------------------------------------
__global__ __launch_bounds__(256) void head_kernel(const float* __restrict__ win,
                                                   const float* __restrict__ wpad,
                                                   const float* __restrict__ bias,
                                                   float* __restrict__ out) {
  const int m0    = blockIdx.x * 16;    // 25 blocks -> rows 0..399
  const int wave  = threadIdx.x >> 5;   // 8 waves -> 8 N-tiles (128 outputs)
  const int lane  = threadIdx.x & 31;
  const int n0    = wave * 16;
  const int lrow  = lane & 15;
  const int khalf = (lane >> 4) * 2;

  const int r = m0 + lrow;              // A row; always < 400
  const int o = n0 + lrow;              // B column (output channel), < 128
  const float* aRow = win + (size_t)r * KPAD + khalf;
  const float* bRow = wpad + (size_t)o * KPAD + khalf;

  v8f c = {};
#pragma unroll 13
  for (int k = 0; k < KPAD; k += 4) {
    const v2f a = *(const v2f*)(aRow + k);
    const v2f w = *(const v2f*)(bRow + k);
    c = __builtin_amdgcn_wmma_f32_16x16x4_f32(false, a, false, w,
                                              (short)0, c, false, false);
  }

  const int nn    = n0 + lrow;
  const int rbase = m0 + ((lane >> 4) * 8);
  const float bv  = bias[nn];
  float* dst = out + (size_t)rbase * OUTF + nn;
#pragma unroll
  for (int i = 0; i < 8; ++i) {
    const float v = c[i] + bv;
    dst[i * OUTF] = v > 0.f ? v : 0.f;
  }
}

// ---------------------------------------------------------------------------
extern "C" void kernel_launch(void* const* d_in, const int* in_sizes, int n_in,
                              void* d_out, int out_size, void* d_ws, size_t ws_size,
                              hipStream_t stream) {
  const int*   frames = (const int*)d_in[0];    // [4,3,100,224,224] int32
  const float* Wm     = (const float*)d_in[1];  // [128,101] f32
  const float* bias   = (const float*)d_in[2];  // [128] f32
  float*       out    = (float*)d_out;          // [4,100,128] f32

  float* x    = (float*)d_ws;                         // [4,112,512]  ~917 KB
  float* sim  = x + (size_t)BB * TPAD * NBINS;        // [4,112,112]  ~200 KB
  float* win  = sim + (size_t)BB * TPAD * TPAD;       // [400,104]    ~166 KB
  float* wpad = win + (size_t)NWIN;                   // [128,104]    ~53 KB

  xpad_kernel<<<(BB * (TPAD - TT) * NBINS + 255) / 256, 256, 0, stream>>>(x);
  hist_kernel<<<BB * TT, 256, 0, stream>>>(frames, x);
  sim_kernel<<<dim3(7, BB), 224, 0, stream>>>(x, sim);
  prep_kernel<<<(NWIN + NWP + 255) / 256, 256, 0, stream>>>(sim, Wm, win, wpad);
  head_kernel<<<25, 256, 0, stream>>>(win, wpad, bias, out);
}